// InteractionBlock_14482629722857
// MI455X (gfx1250) — compile-verified
//
#include <hip/hip_runtime.h>

typedef __attribute__((ext_vector_type(2))) float v2f;
typedef __attribute__((ext_vector_type(8))) float v8f;

#define N_NODES 50000
#define N_EDGES 1600000
#define NAB 128
#define NF 128
#define NG 50

// ssp(x) = softplus(x) - ln(2), numerically stable softplus
__device__ __forceinline__ float ssp_f(float x) {
    float sp = fmaxf(x, 0.0f) + log1pf(expf(-fabsf(x)));
    return sp - 0.69314718055994530942f;
}

__global__ void zero_kernel(float* __restrict__ p, int n) {
    int i = blockIdx.x * blockDim.x + threadIdx.x;
    if (i < n) p[i] = 0.0f;
}

// rf = r (N x 128) @ W_af (128 x 128)
// grid.x = N/16 row strips; 8 waves per block = 8 column tiles of 16
__global__ __launch_bounds__(256) void rf_kernel(const float* __restrict__ r,
                                                 const float* __restrict__ W_af,
                                                 float* __restrict__ rf) {
    const int lane = threadIdx.x & 31;
    const int wave = threadIdx.x >> 5;      // column tile 0..7
    const int m    = lane & 15;
    const int half = lane >> 4;
    const int rowBase = blockIdx.x * 16;
    const int n = wave * 16 + m;

    v8f acc = {};
    const float* arow = r + (size_t)(rowBase + m) * NAB;
    for (int k = 0; k < NAB; k += 4) {
        const int ka = k + 2 * half;
        v2f af = *(const v2f*)(arow + ka);           // A: M=lane%16, K pair
        v2f bf;
        bf.x = W_af[(size_t)ka * NF + n];            // B: K rows striped across halves
        bf.y = W_af[(size_t)(ka + 1) * NF + n];
        acc = __builtin_amdgcn_wmma_f32_16x16x4_f32(false, af, false, bf,
                                                    (short)0, acc, false, false);
    }
#pragma unroll
    for (int j = 0; j < 8; ++j)
        rf[(size_t)(rowBase + j + 8 * half) * NF + n] = acc[j];
}

#define EDGE_WAVES 8

// Fully fused per-edge pipeline: gaussians -> MLP(50->50->128) via WMMA ->
// gather rf -> scatter-add into agg. One wave handles 16 edges.
__global__ __launch_bounds__(256) void edge_kernel(
    const float* __restrict__ e,
    const int*   __restrict__ a,
    const float* __restrict__ W_df1, const float* __restrict__ b_df1,
    const float* __restrict__ W_df2, const float* __restrict__ b_df2,
    const float* __restrict__ rf,
    float* __restrict__ agg)
{
    __shared__ float hbuf[EDGE_WAVES][16][66];   // 66 pad: conflict-free b64 transpose reads

    const int lane = threadIdx.x & 31;
    const int wave = threadIdx.x >> 5;
    const int m    = lane & 15;
    const int half = lane >> 4;
    const int tile = blockIdx.x * EDGE_WAVES + wave;   // E/16 tiles exactly
    const int base = tile * 16;

    const float em   = e[base + m];          // edge distance for row m
    const int   sidx = a[2 * (base + m) + 0];
    const int   didx = a[2 * (base + m) + 1];

    const float width = 5.0f / 49.0f;
    const float coeff = -0.5f / (width * width);

    // ---- GEMM1: h_pre = g(16 x 50) @ W_df1(50 x 50); K padded to 52, N to 64 ----
    v8f acc1[4] = {};
    for (int kk = 0; kk < 13; ++kk) {
        const int ka = kk * 4 + 2 * half;
        // Gaussian basis generated directly in WMMA A-fragment layout
        v2f af;
        {
            float d0 = em - (float)ka * width;
            float d1 = em - (float)(ka + 1) * width;
            af.x = (ka     < NG) ? expf(coeff * d0 * d0) : 0.0f;
            af.y = (ka + 1 < NG) ? expf(coeff * d1 * d1) : 0.0f;
        }
#pragma unroll
        for (int t = 0; t < 4; ++t) {
            const int n = t * 16 + m;
            v2f bf;
            bf.x = (ka     < NG && n < NG) ? W_df1[ka * NG + n]       : 0.0f;
            bf.y = (ka + 1 < NG && n < NG) ? W_df1[(ka + 1) * NG + n] : 0.0f;
            acc1[t] = __builtin_amdgcn_wmma_f32_16x16x4_f32(false, af, false, bf,
                                                            (short)0, acc1[t], false, false);
        }
    }

    // bias + ssp, stage to LDS for the A-layout transpose
#pragma unroll
    for (int t = 0; t < 4; ++t) {
        const int col = t * 16 + m;
        const float bias = (col < NG) ? b_df1[col] : 0.0f;
#pragma unroll
        for (int j = 0; j < 8; ++j)
            hbuf[wave][j + 8 * half][col] = ssp_f(acc1[t][j] + bias);
    }
    __syncthreads();   // all waves follow identical control flow

    // ---- GEMM2: ef = h(16 x 52) @ W_df2(50 x 128); rows >= 50 predicated to 0 ----
    v8f acc2[8] = {};
    for (int kk = 0; kk < 13; ++kk) {
        const int ka = kk * 4 + 2 * half;
        v2f af = *(const v2f*)&hbuf[wave][m][ka];
#pragma unroll
        for (int t = 0; t < 8; ++t) {
            const int n = t * 16 + m;
            v2f bf;
            bf.x = (ka     < NG) ? W_df2[ka * NF + n]       : 0.0f;
            bf.y = (ka + 1 < NG) ? W_df2[(ka + 1) * NF + n] : 0.0f;
            acc2[t] = __builtin_amdgcn_wmma_f32_16x16x4_f32(false, af, false, bf,
                                                            (short)0, acc2[t], false, false);
        }
    }

    // ---- epilogue: ef = acc2 + b2; gather rf[src]/rf[dst]; scatter-add ----
    float b2[8];
#pragma unroll
    for (int t = 0; t < 8; ++t) b2[t] = b_df2[t * 16 + m];

#pragma unroll
    for (int j = 0; j < 8; ++j) {
        const int ej = j + 8 * half;                  // edge row for this acc vgpr
        const int s  = __shfl(sidx, ej, 32);
        const int d  = __shfl(didx, ej, 32);
        const float* rs = rf + (size_t)s * NF;
        const float* rd = rf + (size_t)d * NF;
        float* as = agg + (size_t)s * NF;
        float* ad = agg + (size_t)d * NF;
#pragma unroll
        for (int t = 0; t < 8; ++t) {
            const int feat = t * 16 + m;
            const float efv = acc2[t][j] + b2[t];
            unsafeAtomicAdd(&ad[feat], rs[feat] * efv);   // segment_sum(m1, dst)
            unsafeAtomicAdd(&as[feat], rd[feat] * efv);   // segment_sum(m2, src)
        }
    }
}

#define OUT_WAVES 4

// out = ssp(agg @ W_d1 + b_d1) @ W_d2 + b_d2 ; one wave = one 16-row strip
__global__ __launch_bounds__(128) void out_kernel(
    const float* __restrict__ agg,
    const float* __restrict__ W_d1, const float* __restrict__ b_d1,
    const float* __restrict__ W_d2, const float* __restrict__ b_d2,
    float* __restrict__ out)
{
    __shared__ float hbuf[OUT_WAVES][16][130];

    const int lane = threadIdx.x & 31;
    const int wave = threadIdx.x >> 5;
    const int m    = lane & 15;
    const int half = lane >> 4;
    const int strip  = blockIdx.x * OUT_WAVES + wave;
    const bool active = strip < (N_NODES / 16);
    const int base = (active ? strip : 0) * 16;   // clamp; keep control flow uniform

    // ---- GEMM1: agg(16x128) @ W_d1(128x128) ----
    v8f acc1[8] = {};
    const float* arow = agg + (size_t)(base + m) * NF;
    for (int k = 0; k < NF; k += 4) {
        const int ka = k + 2 * half;
        v2f af = *(const v2f*)(arow + ka);
#pragma unroll
        for (int t = 0; t < 8; ++t) {
            const int n = t * 16 + m;
            v2f bf;
            bf.x = W_d1[(size_t)ka * NAB + n];
            bf.y = W_d1[(size_t)(ka + 1) * NAB + n];
            acc1[t] = __builtin_amdgcn_wmma_f32_16x16x4_f32(false, af, false, bf,
                                                            (short)0, acc1[t], false, false);
        }
    }
#pragma unroll
    for (int t = 0; t < 8; ++t) {
        const int col = t * 16 + m;
        const float bias = b_d1[col];
#pragma unroll
        for (int j = 0; j < 8; ++j)
            hbuf[wave][j + 8 * half][col] = ssp_f(acc1[t][j] + bias);
    }
    __syncthreads();

    // ---- GEMM2: h(16x128) @ W_d2(128x128) ----
    v8f acc2[8] = {};
    for (int k = 0; k < NAB; k += 4) {
        const int ka = k + 2 * half;
        v2f af = *(const v2f*)&hbuf[wave][m][ka];
#pragma unroll
        for (int t = 0; t < 8; ++t) {
            const int n = t * 16 + m;
            v2f bf;
            bf.x = W_d2[(size_t)ka * NAB + n];
            bf.y = W_d2[(size_t)(ka + 1) * NAB + n];
            acc2[t] = __builtin_amdgcn_wmma_f32_16x16x4_f32(false, af, false, bf,
                                                            (short)0, acc2[t], false, false);
        }
    }
    if (active) {
#pragma unroll
        for (int t = 0; t < 8; ++t) {
            const int n = t * 16 + m;
            const float bias = b_d2[n];
#pragma unroll
            for (int j = 0; j < 8; ++j)
                out[(size_t)(base + j + 8 * half) * NAB + n] = acc2[t][j] + bias;
        }
    }
}

extern "C" void kernel_launch(void* const* d_in, const int* in_sizes, int n_in,
                              void* d_out, int out_size, void* d_ws, size_t ws_size,
                              hipStream_t stream) {
    const float* r     = (const float*)d_in[0];
    const float* e     = (const float*)d_in[1];
    const int*   a     = (const int*)d_in[2];
    const float* W_df1 = (const float*)d_in[3];
    const float* b_df1 = (const float*)d_in[4];
    const float* W_df2 = (const float*)d_in[5];
    const float* b_df2 = (const float*)d_in[6];
    const float* W_af  = (const float*)d_in[7];
    const float* W_d1  = (const float*)d_in[8];
    const float* b_d1  = (const float*)d_in[9];
    const float* W_d2  = (const float*)d_in[10];
    const float* b_d2  = (const float*)d_in[11];
    float* out = (float*)d_out;

    float* rf  = (float*)d_ws;                       // N x 128
    float* agg = rf + (size_t)N_NODES * NF;          // N x 128

    const int aggN = N_NODES * NF;
    zero_kernel<<<(aggN + 255) / 256, 256, 0, stream>>>(agg, aggN);

    rf_kernel<<<N_NODES / 16, 256, 0, stream>>>(r, W_af, rf);

    edge_kernel<<<N_EDGES / 16 / EDGE_WAVES, 256, 0, stream>>>(
        e, a, W_df1, b_df1, W_df2, b_df2, rf, agg);

    out_kernel<<<(N_NODES / 16 + OUT_WAVES - 1) / OUT_WAVES, 32 * OUT_WAVES, 0, stream>>>(
        agg, W_d1, b_d1, W_d2, b_d2, out);
}